// ForwardBPTTCell_12403865551495
// MI455X (gfx1250) — compile-verified
//
#include <hip/hip_runtime.h>
#include <math.h>

// ForwardBPTTCell on MI455X (gfx1250, wave32).
// B=1024, H=256, O=128, D=64. All data f32.
// GEMMs use V_WMMA_F32_16X16X4_F32 (full-precision matrix pipe).
// The batched 256x256x256 matmul (34 GFLOP hot spot) stages K-chunks into LDS
// with the Tensor Data Mover (tensor_load_to_lds, TENSORcnt) double-buffered,
// using TDM pad_enable to land tiles at pitch 33/65 (bank-conflict free).
// Batched 256x256 solve runs one workgroup per batch in 265KB of LDS
// (320KB/WGP on CDNA5) via Gauss-Jordan, 257-float pitch.

#define BDIM 1024
#define HDIM 256
#define ODIM 128
#define DDIM 64
#define BH   (BDIM*HDIM)

typedef __attribute__((ext_vector_type(2))) float v2f;
typedef __attribute__((ext_vector_type(8))) float v8f;
typedef __attribute__((ext_vector_type(4))) unsigned int u32x4;
typedef __attribute__((ext_vector_type(4))) int i32x4;
typedef __attribute__((ext_vector_type(8))) int i32x8;

// ---------------- WMMA helpers (f32 16x16x4) ----------------
__device__ __forceinline__ v8f wmma4(v2f a, v2f b, v8f c) {
  return __builtin_amdgcn_wmma_f32_16x16x4_f32(
      /*neg_a=*/false, a, /*neg_b=*/false, b,
      /*c_mod=*/(short)0, c, /*reuse_a=*/false, /*reuse_b=*/false);
}

// A fragment: 16x4 tile, rows r0..r0+15, cols k0..k0+3 of row-major M (stride ld).
// ISA layout: lanes 0-15 hold K={k0,k0+1}, lanes 16-31 hold K={k0+2,k0+3}.
__device__ __forceinline__ v2f ldA(const float* __restrict__ M, int ld, int r0, int k0, int lane) {
  int m = r0 + (lane & 15);
  int k = k0 + ((lane >> 4) << 1);
  v2f a; a.x = M[m * ld + k]; a.y = M[m * ld + k + 1]; return a;
}
// B fragment: 4x16 tile, rows k0..k0+3, cols n0..n0+15 of row-major M (stride ld).
__device__ __forceinline__ v2f ldB(const float* __restrict__ M, int ld, int k0, int n0, int lane) {
  int n = n0 + (lane & 15);
  int k = k0 + ((lane >> 4) << 1);
  v2f b; b.x = M[k * ld + n]; b.y = M[(k + 1) * ld + n]; return b;
}
// B fragment reading M transposed: B[k][n] = M[n][k] (M row-major, stride ld).
__device__ __forceinline__ v2f ldBT(const float* __restrict__ M, int ld, int k0, int n0, int lane) {
  int n = n0 + (lane & 15);
  int k = k0 + ((lane >> 4) << 1);
  v2f b; b.x = M[n * ld + k]; b.y = M[n * ld + k + 1]; return b;
}
// C/D store: VGPR v -> row r0+v (lanes 0-15) / r0+8+v (lanes 16-31), col n0+(lane&15).
__device__ __forceinline__ void stC(float* __restrict__ D, int ld, int r0, int c0, int lane, v8f c) {
  int n  = c0 + (lane & 15);
  int m0 = r0 + ((lane >> 4) << 3);
#pragma unroll
  for (int v = 0; v < 8; ++v) D[(m0 + v) * ld + n] = c[v];
}

// ---------------- TDM: tensor_load_to_lds of a 2D f32 tile ----------------
// D# per CDNA5 ISA. Tensor is 256x256 f32, row stride 256.
// g1w0 packs data_size=4B | pad_enable | pad_interval (pad 1 dword per tile row)
// so the tile lands in LDS with pitch tile_d0+1 -> conflict-free column reads.
__device__ __forceinline__ void tdm_load_tile_f32(
    unsigned long long gaddr, unsigned lds_off, unsigned g1w0,
    unsigned tile_d0, unsigned tile_d1) {
  u32x4 g0;
  g0[0] = 1u;                                               // count=1, user D#
  g0[1] = lds_off;                                          // lds_addr (bytes)
  g0[2] = (unsigned)(gaddr & 0xffffffffull);                // global_addr[31:0]
  g0[3] = (unsigned)((gaddr >> 32) & 0x01ffffffull)         // global_addr[56:32]
          | 0x80000000u;                                    // type=2 ("image")
  i32x8 g1;
  g1[0] = (int)g1w0;                    // wg_mask=0 | data_size=2(4B) | pad cfg
  g1[1] = (int)(256u << 16);            // atomic_barrier=0 | tensor_dim0 lo16
  g1[2] = (int)(256u << 16);            // tensor_dim0 hi16=0 | tensor_dim1 lo16
  g1[3] = (int)(tile_d0 << 16);         // tensor_dim1 hi16=0 | tile_dim0
  g1[4] = (int)tile_d1;                 // tile_dim1 | tile_dim2=0
  g1[5] = 256;                          // tensor_dim0_stride[31:0]
  g1[6] = 0;                            // stride0 hi | tensor_dim1_stride lo
  g1[7] = 0;
  i32x4 gz = {0, 0, 0, 0};              // groups 2/3: dims 2..4 unused
#if defined(__clang_major__) && (__clang_major__ >= 23)
  i32x8 gz8 = {0, 0, 0, 0, 0, 0, 0, 0};
  __builtin_amdgcn_tensor_load_to_lds(g0, g1, gz, gz, gz8, 0);
#else
  __builtin_amdgcn_tensor_load_to_lds(g0, g1, gz, gz, 0);
#endif
}

// ---------------- 1. GRU gate pre-activations (WMMA) ----------------
__global__ __launch_bounds__(128) void gates_wmma(
    const float* __restrict__ x, const float* __restrict__ h,
    const float* __restrict__ Wir, const float* __restrict__ bir, const float* __restrict__ Whr,
    const float* __restrict__ Wiz, const float* __restrict__ biz, const float* __restrict__ Whz,
    const float* __restrict__ Win, const float* __restrict__ bin_,
    const float* __restrict__ Whn, const float* __restrict__ bhn,
    float* __restrict__ pre_r, float* __restrict__ pre_z,
    float* __restrict__ xn, float* __restrict__ gbuf) {
  int lane = threadIdx.x & 31;
  int wave = threadIdx.x >> 5;
  int r0 = blockIdx.x * 16;
  int n0 = blockIdx.y * 16;
  const float* Wi = nullptr; const float* Wh = nullptr; const float* bi = nullptr;
  float* dst = nullptr;
  if (wave == 0)      { Wi = Wir; Wh = Whr; bi = bir;  dst = pre_r; }
  else if (wave == 1) { Wi = Wiz; Wh = Whz; bi = biz;  dst = pre_z; }
  else if (wave == 2) { Wi = Win;           bi = bin_; dst = xn;    }
  else                {           Wh = Whn; bi = bhn;  dst = gbuf;  }
  v8f acc = {0.f,0.f,0.f,0.f,0.f,0.f,0.f,0.f};
  if (Wi)
    for (int k0 = 0; k0 < DDIM; k0 += 4)
      acc = wmma4(ldA(x, DDIM, r0, k0, lane), ldB(Wi, HDIM, k0, n0, lane), acc);
  if (Wh)
    for (int k0 = 0; k0 < HDIM; k0 += 4)
      acc = wmma4(ldA(h, HDIM, r0, k0, lane), ldB(Wh, HDIM, k0, n0, lane), acc);
  float bv = bi[n0 + (lane & 15)];
#pragma unroll
  for (int v = 0; v < 8; ++v) acc[v] += bv;
  stC(dst, HDIM, r0, n0, lane, acc);
}

// ---------------- 2. GRU pointwise + jacobian coefficients ----------------
__global__ __launch_bounds__(256) void gru_pointwise(
    const float* __restrict__ pre_r, const float* __restrict__ pre_z,
    const float* __restrict__ xn, const float* __restrict__ gbuf,
    const float* __restrict__ h, const float* __restrict__ delta,
    const float* __restrict__ inst_delta, const float* __restrict__ t,
    float* __restrict__ new_h, float* __restrict__ new_mean, float* __restrict__ h_out,
    float* __restrict__ new_t, float* __restrict__ zb, float* __restrict__ czb,
    float* __restrict__ cnb, float* __restrict__ crb, float* __restrict__ rhsb) {
  int idx = blockIdx.x * 256 + threadIdx.x;
  if (idx >= BH) return;
  float r  = 1.0f / (1.0f + expf(-pre_r[idx]));
  float z  = 1.0f / (1.0f + expf(-pre_z[idx]));
  float gv = gbuf[idx];
  float nv = tanhf(xn[idx] + r * gv);
  float hv = h[idx];
  float nh = (1.0f - z) * nv + z * hv;
  new_h[idx] = nh; new_mean[idx] = nh; h_out[idx] = hv;
  float dz = z * (1.0f - z), dr = r * (1.0f - r), dn = 1.0f - nv * nv;
  czb[idx] = (hv - nv) * dz;
  cnb[idx] = (1.0f - z) * dn * r;
  crb[idx] = (1.0f - z) * dn * gv * dr;
  zb[idx]  = z;
  rhsb[idx] = delta[idx] - inst_delta[idx];
  if (idx < BDIM) new_t[idx] = t[idx] + 1.0f;
}

// ---------------- 3. LayerNorm forward (+ h_norm) ----------------
__global__ __launch_bounds__(256) void ln_forward(
    const float* __restrict__ new_h, const float* __restrict__ scale,
    const float* __restrict__ bias, float* __restrict__ ln_h,
    float* __restrict__ xhat, float* __restrict__ rsb, float* __restrict__ h_norm) {
  __shared__ float red[HDIM];
  int b = blockIdx.x, tid = threadIdx.x;
  float v = new_h[b * HDIM + tid];
  red[tid] = v; __syncthreads();
  for (int s = 128; s > 0; s >>= 1) { if (tid < s) red[tid] += red[tid + s]; __syncthreads(); }
  float mu = red[0] * (1.0f / HDIM); __syncthreads();
  float d = v - mu;
  red[tid] = d * d; __syncthreads();
  for (int s = 128; s > 0; s >>= 1) { if (tid < s) red[tid] += red[tid + s]; __syncthreads(); }
  float var = red[0] * (1.0f / HDIM); __syncthreads();
  float rs = rsqrtf(var + 1e-6f);
  float xh = d * rs;
  ln_h[b * HDIM + tid] = xh * scale[tid] + bias[tid];
  xhat[b * HDIM + tid] = xh;
  if (tid == 0) rsb[b] = rs;
  red[tid] = v * v; __syncthreads();
  for (int s = 128; s > 0; s >>= 1) { if (tid < s) red[tid] += red[tid + s]; __syncthreads(); }
  if (tid == 0) h_norm[b] = sqrtf(red[0]);
}

// ---------------- 4. Readout GEMM (WMMA): out = ln_h @ Wout + bout ----------------
__global__ __launch_bounds__(32) void readout_wmma(
    const float* __restrict__ ln_h, const float* __restrict__ Wout,
    const float* __restrict__ bout, float* __restrict__ out) {
  int lane = threadIdx.x;
  int r0 = blockIdx.x * 16, n0 = blockIdx.y * 16;
  v8f acc = {0.f,0.f,0.f,0.f,0.f,0.f,0.f,0.f};
  for (int k0 = 0; k0 < HDIM; k0 += 4)
    acc = wmma4(ldA(ln_h, HDIM, r0, k0, lane), ldB(Wout, ODIM, k0, n0, lane), acc);
  float bv = bout[n0 + (lane & 15)];
#pragma unroll
  for (int v = 0; v < 8; ++v) acc[v] += bv;
  stC(out, ODIM, r0, n0, lane, acc);
}

// ---------------- 5. delta_output = 2*m*(out-y) ----------------
__global__ __launch_bounds__(256) void delta_out_k(
    const float* __restrict__ out, const float* __restrict__ y,
    const float* __restrict__ m, float* __restrict__ dout) {
  int idx = blockIdx.x * 256 + threadIdx.x;
  if (idx < BDIM * ODIM) dout[idx] = 2.0f * m[idx] * (out[idx] - y[idx]);
}

// ---------------- 6. gh = delta_output @ Wout^T (WMMA, transposed B) ----------------
__global__ __launch_bounds__(32) void backout_wmma(
    const float* __restrict__ dout, const float* __restrict__ Wout,
    float* __restrict__ gh) {
  int lane = threadIdx.x;
  int r0 = blockIdx.x * 16, n0 = blockIdx.y * 16;
  v8f acc = {0.f,0.f,0.f,0.f,0.f,0.f,0.f,0.f};
  for (int k0 = 0; k0 < ODIM; k0 += 4)
    acc = wmma4(ldA(dout, ODIM, r0, k0, lane), ldBT(Wout, ODIM, k0, n0, lane), acc);
  stC(gh, HDIM, r0, n0, lane, acc);
}

// ---------------- 7. LayerNorm backward -> new_inst_delta ----------------
__global__ __launch_bounds__(256) void ln_backward(
    const float* __restrict__ gh, const float* __restrict__ scale,
    const float* __restrict__ xhat, const float* __restrict__ rsb,
    float* __restrict__ new_inst_delta) {
  __shared__ float red[HDIM];
  int b = blockIdx.x, tid = threadIdx.x;
  float gx = gh[b * HDIM + tid] * scale[tid];
  float xh = xhat[b * HDIM + tid];
  red[tid] = gx; __syncthreads();
  for (int s = 128; s > 0; s >>= 1) { if (tid < s) red[tid] += red[tid + s]; __syncthreads(); }
  float mgx = red[0] * (1.0f / HDIM); __syncthreads();
  red[tid] = gx * xh; __syncthreads();
  for (int s = 128; s > 0; s >>= 1) { if (tid < s) red[tid] += red[tid + s]; __syncthreads(); }
  float mgxx = red[0] * (1.0f / HDIM);
  new_inst_delta[b * HDIM + tid] = rsb[b] * (gx - mgx - xh * mgxx);
}

// ---------------- 8. Jacobian build ----------------
// jac[b,i,j] = Whz[i,j]*cz[b,j] + Whn[i,j]*cn[b,j] + Whr[i,j]*cr[b,j] + (i==j)*z[b,j]
__global__ __launch_bounds__(256) void jac_build(
    const float* __restrict__ Whz, const float* __restrict__ Whn, const float* __restrict__ Whr,
    const float* __restrict__ czb, const float* __restrict__ cnb, const float* __restrict__ crb,
    const float* __restrict__ zb, float* __restrict__ jac) {
  int idx  = blockIdx.x * 256 + threadIdx.x;  // < B*H*H = 2^26
  int j    = idx & (HDIM - 1);
  int rest = idx >> 8;
  int i    = rest & (HDIM - 1);
  int b    = rest >> 8;
  int w    = i * HDIM + j;
  int bj   = b * HDIM + j;
  float v = Whz[w] * czb[bj] + Whn[w] * cnb[bj] + Whr[w] * crb[bj];
  if (i == j) v += zb[bj];
  jac[idx] = v;
}

// ---------------- 9. Batched GEMM: new_prod_jac[b] = prod_jac[b] @ jac[b] ----------------
// 8 waves per workgroup compute a 64x64 block; K staged in 32-wide chunks into
// LDS by the TDM (double-buffered, TENSORcnt), each wave does 16x32 (2 accums).
#define KC   32                    // K chunk
#define APITCH 33                  // TDM pad: +1 dword per 32-dword row
#define BPITCH 65                  // TDM pad: +1 dword per 64-dword row
__global__ __launch_bounds__(256) void bmm_wmma(
    const float* __restrict__ prod_jac, const float* __restrict__ jac,
    float* __restrict__ new_prod_jac) {
  __shared__ float Alds[2][64 * APITCH];   // 64 rows x 32 cols (pitch 33)
  __shared__ float Blds[2][KC * BPITCH];   // 32 rows x 64 cols (pitch 65)
  int lane = threadIdx.x & 31;
  int wave = threadIdx.x >> 5;
  int rg = wave >> 1;            // 0..3: 16-row group inside 64-row block
  int cg = wave & 1;             // 0..1: 32-col group inside 64-col block
  int b  = blockIdx.z;
  int rowBase = blockIdx.y * 64;
  int colBase = blockIdx.x * 64;
  unsigned long long Ag = (unsigned long long)(const void*)(prod_jac + (size_t)b * HDIM * HDIM);
  unsigned long long Bg = (unsigned long long)(const void*)(jac      + (size_t)b * HDIM * HDIM);
  const unsigned g1A = 0x20000u | (1u << 20) | (4u << 22); // 4B, pad 1dw / 32dw
  const unsigned g1B = 0x20000u | (1u << 20) | (5u << 22); // 4B, pad 1dw / 64dw

  int m  = rg * 16 + (lane & 15);            // A row in 64-row block
  int n0 = cg * 32 + (lane & 15);            // B cols for the two accumulators
  int n1 = n0 + 16;
  v8f acc0 = {0.f,0.f,0.f,0.f,0.f,0.f,0.f,0.f};
  v8f acc1 = {0.f,0.f,0.f,0.f,0.f,0.f,0.f,0.f};

  // prologue: chunk 0 -> buffer 0 (wave 0 drives the TDM; EXEC is ignored by TDM)
  if (wave == 0) {
    tdm_load_tile_f32(Ag + (unsigned long long)(rowBase * HDIM) * 4u,
                      (unsigned)(unsigned long long)(const void*)&Alds[0][0], g1A, KC, 64);
    tdm_load_tile_f32(Bg + (unsigned long long)(colBase) * 4u,
                      (unsigned)(unsigned long long)(const void*)&Blds[0][0], g1B, 64, KC);
  }
#pragma unroll 1
  for (int c = 0; c < HDIM / KC; ++c) {
    int buf = c & 1;
    if (wave == 0) {
      if (c < HDIM / KC - 1) {
        int k0n = (c + 1) * KC;
        tdm_load_tile_f32(Ag + (unsigned long long)(rowBase * HDIM + k0n) * 4u,
                          (unsigned)(unsigned long long)(const void*)&Alds[buf ^ 1][0], g1A, KC, 64);
        tdm_load_tile_f32(Bg + (unsigned long long)(k0n * HDIM + colBase) * 4u,
                          (unsigned)(unsigned long long)(const void*)&Blds[buf ^ 1][0], g1B, 64, KC);
        __builtin_amdgcn_s_wait_tensorcnt(2);   // current chunk's 2 TDM ops done
      } else {
        __builtin_amdgcn_s_wait_tensorcnt(0);
      }
    }
    __syncthreads();
    const float* Ab = &Alds[buf][0];
    const float* Bb = &Blds[buf][0];
#pragma unroll
    for (int kl = 0; kl < KC; kl += 4) {
      int kk = kl + ((lane >> 4) << 1);
      v2f aF; aF.x = Ab[m * APITCH + kk];  aF.y = Ab[m * APITCH + kk + 1];
      v2f b0; b0.x = Bb[kk * BPITCH + n0]; b0.y = Bb[(kk + 1) * BPITCH + n0];
      v2f b1; b1.x = Bb[kk * BPITCH + n1]; b1.y = Bb[(kk + 1) * BPITCH + n1];
      acc0 = wmma4(aF, b0, acc0);
      acc1 = wmma4(aF, b1, acc1);
    }
    __syncthreads();   // protect buffers before next TDM overwrite
  }
  float* D = new_prod_jac + (size_t)b * HDIM * HDIM;
  stC(D, HDIM, rowBase + rg * 16, colBase + cg * 32,      lane, acc0);
  stC(D, HDIM, rowBase + rg * 16, colBase + cg * 32 + 16, lane, acc1);
}

// ---------------- 10. prod_jac_norm ----------------
__global__ __launch_bounds__(256) void pjnorm(
    const float* __restrict__ npj, float* __restrict__ prod_jac_norm) {
  __shared__ float red[HDIM];
  int b = blockIdx.x, tid = threadIdx.x;
  const float* P = npj + (size_t)b * HDIM * HDIM;
  float s = 0.0f;
  for (int i = tid; i < HDIM * HDIM; i += 256) { float v = P[i]; s += v * v; }
  red[tid] = s; __syncthreads();
  for (int st = 128; st > 0; st >>= 1) { if (tid < st) red[tid] += red[tid + st]; __syncthreads(); }
  if (tid == 0) prod_jac_norm[b] = sqrtf(red[0]);
}

// ---------------- 11. Batched 256x256 solve in LDS (Gauss-Jordan) ----------------
__global__ __launch_bounds__(256, 1) void solve_kernel(
    const float* __restrict__ jac, const float* __restrict__ rhsb,
    float* __restrict__ new_delta, float* __restrict__ delta_norm) {
  __shared__ float As[HDIM][HDIM + 1];  // pitch 257 -> (i+j)%64 bank pattern
  __shared__ float rv[HDIM];
  __shared__ float fv[HDIM];
  int b = blockIdx.x, tid = threadIdx.x;
  const float* J = jac + (size_t)b * HDIM * HDIM;
  for (int i = 0; i < HDIM; ++i) As[i][tid] = J[i * HDIM + tid];  // coalesced
  rv[tid] = rhsb[b * HDIM + tid];
  for (int k = 0; k < HDIM; ++k) {
    __syncthreads();
    float pivinv = 1.0f / As[k][k];                    // broadcast read
    fv[tid] = (tid == k) ? 0.0f : As[tid][k] * pivinv; // column read, conflict-free
    __syncthreads();
    if (tid > k) {            // thread owns column tid
      float akj = As[k][tid];
#pragma unroll 4
      for (int i = 0; i < HDIM; ++i) As[i][tid] -= fv[i] * akj;
    } else if (tid == k) {    // freed thread updates the RHS column
      float rk = rv[k];
#pragma unroll 4
      for (int i = 0; i < HDIM; ++i) rv[i] -= fv[i] * rk;
    }
  }
  __syncthreads();
  float xv = rv[tid] / As[tid][tid];
  new_delta[b * HDIM + tid] = xv;
  fv[tid] = xv * xv; __syncthreads();
  for (int s = 128; s > 0; s >>= 1) { if (tid < s) fv[tid] += fv[tid + s]; __syncthreads(); }
  if (tid == 0) delta_norm[b] = sqrtf(fv[0]);
}

// ---------------- host launch ----------------
extern "C" void kernel_launch(void* const* d_in, const int* in_sizes, int n_in,
                              void* d_out, int out_size, void* d_ws, size_t ws_size,
                              hipStream_t stream) {
  const float* x          = (const float*)d_in[0];
  const float* y          = (const float*)d_in[1];
  const float* m          = (const float*)d_in[2];
  const float* h          = (const float*)d_in[3];
  const float* delta      = (const float*)d_in[4];
  const float* inst_delta = (const float*)d_in[5];
  const float* prod_jac   = (const float*)d_in[6];
  /* prev_mean d_in[7] unused (pooling == 'none') */
  const float* t          = (const float*)d_in[8];
  const float* Wiz = (const float*)d_in[9];
  const float* biz = (const float*)d_in[10];
  const float* Whz = (const float*)d_in[11];
  const float* Wir = (const float*)d_in[12];
  const float* bir = (const float*)d_in[13];
  const float* Whr = (const float*)d_in[14];
  const float* Win = (const float*)d_in[15];
  const float* bin_= (const float*)d_in[16];
  const float* Whn = (const float*)d_in[17];
  const float* bhn = (const float*)d_in[18];
  const float* ln_scale = (const float*)d_in[19];
  const float* ln_bias  = (const float*)d_in[20];
  const float* Wout     = (const float*)d_in[21];
  const float* bout     = (const float*)d_in[22];

  // outputs, flat in return order
  float* o = (float*)d_out;
  float* new_h          = o;                       // B*H
  float* new_delta      = o + 262144;              // B*H
  float* new_inst_delta = o + 524288;              // B*H
  float* new_prod_jac   = o + 786432;              // B*H*H
  float* new_mean       = o + 67895296;            // B*H
  float* new_t          = o + 68157440;            // B
  float* out            = o + 68158464;            // B*O
  float* h_out          = o + 68289536;            // B*H (input h passthrough)
  float* delta_output   = o + 68551680;            // B*O
  float* h_norm         = o + 68682752;            // B
  float* delta_norm     = o + 68683776;            // B
  float* prod_jac_norm  = o + 68684800;            // B
  float* jac            = o + 68685824;            // B*H*H

  // workspace
  float* w = (float*)d_ws;
  float* pre_r = w;            float* pre_z = w + 1*BH;
  float* xn    = w + 2*BH;     float* gbuf  = w + 3*BH;
  float* zb    = w + 4*BH;     float* czb   = w + 5*BH;
  float* cnb   = w + 6*BH;     float* crb   = w + 7*BH;
  float* ln_h  = w + 8*BH;     float* xhat  = w + 9*BH;
  float* gh    = w + 10*BH;    float* rhsb  = w + 11*BH;
  float* rsb   = w + 12*BH;    // B floats

  gates_wmma<<<dim3(BDIM/16, HDIM/16), 128, 0, stream>>>(
      x, h, Wir, bir, Whr, Wiz, biz, Whz, Win, bin_, Whn, bhn,
      pre_r, pre_z, xn, gbuf);
  gru_pointwise<<<BH/256, 256, 0, stream>>>(
      pre_r, pre_z, xn, gbuf, h, delta, inst_delta, t,
      new_h, new_mean, h_out, new_t, zb, czb, cnb, crb, rhsb);
  ln_forward<<<BDIM, 256, 0, stream>>>(new_h, ln_scale, ln_bias, ln_h, xhat, rsb, h_norm);
  readout_wmma<<<dim3(BDIM/16, ODIM/16), 32, 0, stream>>>(ln_h, Wout, bout, out);
  delta_out_k<<<(BDIM*ODIM)/256, 256, 0, stream>>>(out, y, m, delta_output);
  backout_wmma<<<dim3(BDIM/16, HDIM/16), 32, 0, stream>>>(delta_output, Wout, gh);
  ln_backward<<<BDIM, 256, 0, stream>>>(gh, ln_scale, xhat, rsb, new_inst_delta);
  jac_build<<<(BDIM*HDIM*HDIM)/256, 256, 0, stream>>>(Whz, Whn, Whr, czb, cnb, crb, zb, jac);
  bmm_wmma<<<dim3(HDIM/64, HDIM/64, BDIM), 256, 0, stream>>>(prod_jac, jac, new_prod_jac);
  pjnorm<<<BDIM, 256, 0, stream>>>(new_prod_jac, prod_jac_norm);
  solve_kernel<<<BDIM, 256, 0, stream>>>(jac, rhsb, new_delta, delta_norm);
}